// VQVAE_6708738916534
// MI455X (gfx1250) — compile-verified
//
#include <hip/hip_runtime.h>
#include <hip/hip_bf16.h>
#include <math.h>

// ---------------------------------------------------------------------------
// VQ-VAE quantization for MI455X (gfx1250), wave32 + V_WMMA_F32_16X16X4_F32.
//
// x:         [8, 64, 32, 32, 32] f32   (B, C, D, H, W)
// embedding: [512, 64] f32
// out:       [ loss(1) | quantized [8,64,32,32,32] (16777216) | perplexity(1) ]
// ---------------------------------------------------------------------------

typedef __attribute__((ext_vector_type(2))) float v2f;
typedef __attribute__((ext_vector_type(8))) float v8f;

#define NUM_EMB      512
#define EMB_DIM      64
#define SPATIAL      32768          // 32*32*32
#define NROWS        (8 * SPATIAL)  // 262144
#define NTILES       (NROWS / 16)   // 16384 row tiles of 16
#define LDS_STRIDE   66             // 64 + 2 pad floats -> bank-conflict-free b64
#define NUMEL        16777216       // 8*64*32768
#define COMMIT       0.25f

// ---------------------------------------------------------------------------
__global__ void vq_init_kernel(unsigned* g_hist, double* g_loss) {
    int i = threadIdx.x;
    if (i < NUM_EMB) g_hist[i] = 0u;
    if (i == 0) *g_loss = 0.0;
}

// ---------------------------------------------------------------------------
__global__ __launch_bounds__(256)
void vq_main_kernel(const float* __restrict__ x,
                    const float* __restrict__ emb,
                    float* __restrict__ out_q,     // d_out + 1
                    unsigned* __restrict__ g_hist,
                    double* __restrict__ g_loss) {
    extern __shared__ char smem_raw[];
    float*    e_s      = (float*)smem_raw;                     // 512*66 f32
    float*    enorm    = e_s + NUM_EMB * LDS_STRIDE;           // 512 f32
    unsigned* hist     = (unsigned*)(enorm + NUM_EMB);         // 512 u32
    float*    blk_loss = (float*)(hist + NUM_EMB);             // 1 f32

    // ---- stage codebook into LDS (padded rows) ----
    for (int i = threadIdx.x; i < NUM_EMB * EMB_DIM; i += blockDim.x) {
        int c = i >> 6, d = i & 63;
        e_s[c * LDS_STRIDE + d] = emb[i];
    }
    for (int i = threadIdx.x; i < NUM_EMB; i += blockDim.x) hist[i] = 0u;
    if (threadIdx.x == 0) *blk_loss = 0.0f;
    __syncthreads();

    // ---- |e|^2 per code ----
    for (int c = threadIdx.x; c < NUM_EMB; c += blockDim.x) {
        const float* row = e_s + c * LDS_STRIDE;
        float s = 0.0f;
        #pragma unroll
        for (int d = 0; d < EMB_DIM; ++d) s += row[d] * row[d];
        enorm[c] = s;
    }
    __syncthreads();

    const int lane = threadIdx.x & 31;
    const int wave = threadIdx.x >> 5;
    const int m = lane & 15;      // row/col within tile (A: M, B: N)
    const int h = lane >> 4;      // K-half selector per ISA 16x4/4x16 layout

    float lpart = 0.0f;

    const int wavesTotal = gridDim.x * (blockDim.x >> 5);
    for (int tile = blockIdx.x * (blockDim.x >> 5) + wave; tile < NTILES;
         tile += wavesTotal) {
        const int r0 = tile * 16;
        const int b  = r0 >> 15;          // / SPATIAL
        const int s0 = r0 & (SPATIAL - 1);
        const float* xb = x + (size_t)b * (EMB_DIM * SPATIAL);
        const int srow = s0 + m;

        // ---- load A fragments: 16 rows x 64 dims, f32 16x4 layout ----
        // lane m = row, VGPR pair = {K, K+1}, lanes 16-31 take K+2,K+3
        v2f a[16];
        #pragma unroll
        for (int k = 0; k < 16; ++k) {
            const int d0 = 4 * k + 2 * h;
            a[k].x = xb[(size_t)d0 * SPATIAL + srow];
            a[k].y = xb[(size_t)(d0 + 1) * SPATIAL + srow];
        }

        // ---- running argmin over 32 code tiles ----
        float best[8];
        int   besti[8];
        #pragma unroll
        for (int i = 0; i < 8; ++i) { best[i] = 3.402823e38f; besti[i] = 0; }

        for (int t = 0; t < 32; ++t) {
            v8f acc = {};
            const int code = t * 16 + m;  // B: lane m = column (N)
            #pragma unroll
            for (int k = 0; k < 16; ++k) {
                const int d0 = 4 * k + 2 * h;
                v2f bf = *(const v2f*)&e_s[code * LDS_STRIDE + d0];
                acc = __builtin_amdgcn_wmma_f32_16x16x4_f32(
                    false, a[k], false, bf, (short)0, acc, false, false);
            }
            const float en = enorm[code];
            #pragma unroll
            for (int i = 0; i < 8; ++i) {
                // acc[i] at lane l = dot(row i + 8*h, code)
                const float sc = en - 2.0f * acc[i];
                if (sc < best[i]) { best[i] = sc; besti[i] = code; }
            }
        }

        // ---- cross-lane argmin within each 16-lane group + idx broadcast ----
        int myidx = 0;
        #pragma unroll
        for (int i = 0; i < 8; ++i) {
            float bv = best[i];
            int   bi = besti[i];
            #pragma unroll
            for (int off = 8; off >= 1; off >>= 1) {
                const float ov = __shfl_xor(bv, off, 16);
                const int   oi = __shfl_xor(bi, off, 16);
                if (ov < bv || (ov == bv && oi < bi)) { bv = ov; bi = oi; }
            }
            // lane 0 now holds argmin for row i; lane 16 for row i+8
            if (lane == 0 || lane == 16) atomicAdd(&hist[bi], 1u);
            const int lo = __shfl(bi, 0, 32);
            const int hi = __shfl(bi, 16, 32);
            if (m == i)     myidx = lo;   // both halves: row index = m
            if (m == i + 8) myidx = hi;
        }

        // ---- gather + straight-through write + loss ----
        float* outb = out_q + (size_t)b * (EMB_DIM * SPATIAL);
        const float* qrow = e_s + myidx * LDS_STRIDE;
        #pragma unroll
        for (int k = 0; k < 16; ++k) {
            const int d0 = 4 * k + 2 * h;
            v2f q = *(const v2f*)&qrow[d0];
            const float t0 = q.x - a[k].x;
            const float t1 = q.y - a[k].y;
            lpart += t0 * t0 + t1 * t1;
            outb[(size_t)d0 * SPATIAL + srow]       = a[k].x + t0;
            outb[(size_t)(d0 + 1) * SPATIAL + srow] = a[k].y + t1;
        }
    }

    // ---- block-level reductions, flush to global ----
    atomicAdd(blk_loss, lpart);
    __syncthreads();
    for (int c = threadIdx.x; c < NUM_EMB; c += blockDim.x) {
        const unsigned v = hist[c];
        if (v) atomicAdd(&g_hist[c], v);
    }
    if (threadIdx.x == 0) atomicAdd(g_loss, (double)(*blk_loss));
}

// ---------------------------------------------------------------------------
__global__ void vq_finalize_kernel(const unsigned* __restrict__ g_hist,
                                   const double* __restrict__ g_loss,
                                   float* __restrict__ out) {
    __shared__ float red[256];
    float s = 0.0f;
    for (int c = threadIdx.x; c < NUM_EMB; c += 256) {
        const float p = (float)g_hist[c] * (1.0f / (float)NROWS);
        s += p * logf(p + 1e-10f);
    }
    red[threadIdx.x] = s;
    __syncthreads();
    for (int o = 128; o > 0; o >>= 1) {
        if (threadIdx.x < o) red[threadIdx.x] += red[threadIdx.x + o];
        __syncthreads();
    }
    if (threadIdx.x == 0) {
        const double mse = *g_loss * (1.0 / (double)NUMEL);
        out[0]          = (float)((1.0 + (double)COMMIT) * mse); // q + 0.25*e
        out[NUMEL + 1]  = expf(-red[0]);                         // perplexity
    }
}

// ---------------------------------------------------------------------------
extern "C" void kernel_launch(void* const* d_in, const int* in_sizes, int n_in,
                              void* d_out, int out_size, void* d_ws, size_t ws_size,
                              hipStream_t stream) {
    const float* x   = (const float*)d_in[0];  // [8,64,32,32,32]
    const float* emb = (const float*)d_in[1];  // [512,64]
    float* out = (float*)d_out;

    unsigned* g_hist = (unsigned*)d_ws;                       // 512 u32
    double*   g_loss = (double*)((char*)d_ws + 2048);         // 1 f64 (8B aligned)

    vq_init_kernel<<<1, 512, 0, stream>>>(g_hist, g_loss);

    const size_t shmem = (size_t)(NUM_EMB * LDS_STRIDE + NUM_EMB) * sizeof(float)
                       + (size_t)NUM_EMB * sizeof(unsigned) + sizeof(float);
    // 512 blocks x 8 waves -> 4096 waves, 4 row-tiles each (grid-stride)
    vq_main_kernel<<<512, 256, shmem, stream>>>(x, emb, out + 1, g_hist, g_loss);

    vq_finalize_kernel<<<1, 256, 0, stream>>>(g_hist, g_loss, out);
}